// DiffusionPseudoLabelingLoss_317827579957
// MI455X (gfx1250) — compile-verified
//
#include <hip/hip_runtime.h>
#include <stdint.h>

typedef float v2f __attribute__((ext_vector_type(2)));
typedef float v8f __attribute__((ext_vector_type(8)));
typedef unsigned int u32x4 __attribute__((ext_vector_type(4)));
typedef int i32x8 __attribute__((ext_vector_type(8)));
typedef int i32x4 __attribute__((ext_vector_type(4)));

#define N_NODES   16384
#define WPR       512          // 32-bit words per adjacency row (16384/32)
#define ADJ_WORDS ((size_t)N_NODES * WPR)
#define ALPHA     0.5f
#define N_ITER    10
#define KSPLIT    4            // waves cooperating on one 16-row tile
#define KWORDS    (WPR / KSPLIT)

// ---------------------------------------------------------------------------
// Init: zero the 32 MiB adjacency bitmask, the label vector, and the output.
// ---------------------------------------------------------------------------
__global__ void init_kernel(uint32_t* __restrict__ adj, float* __restrict__ x0,
                            float* __restrict__ out) {
    size_t stride = (size_t)gridDim.x * blockDim.x;
    size_t tid = (size_t)blockIdx.x * blockDim.x + threadIdx.x;
    for (size_t i = tid; i < ADJ_WORDS; i += stride) adj[i] = 0u;
    for (size_t i = tid; i < N_NODES; i += stride) x0[i] = 0.0f;
    if (tid == 0) out[0] = 0.0f;
}

// ---------------------------------------------------------------------------
// Scatter edges into the bitmask: adj[src][dst] = 1. atomicOr collapses dups.
// ---------------------------------------------------------------------------
__global__ void scatter_edges_kernel(const int* __restrict__ ei,
                                     uint32_t* __restrict__ adj, int e_total) {
    int e = blockIdx.x * blockDim.x + threadIdx.x;
    if (e >= e_total) return;
    int s = ei[e];
    int d = ei[e_total + e];
    atomicOr(&adj[(size_t)s * WPR + (d >> 5)], 1u << (d & 31));
}

// ---------------------------------------------------------------------------
// Set label vector entries to 1.0 (duplicates write the same value: benign).
// ---------------------------------------------------------------------------
__global__ void scatter_labels_kernel(const int* __restrict__ labels,
                                      float* __restrict__ x0, int n_pos) {
    int i = blockIdx.x * blockDim.x + threadIdx.x;
    if (i < n_pos) x0[labels[i]] = 1.0f;
}

// ---------------------------------------------------------------------------
// Row degrees via popcount; dinv = deg^-0.5 (self-loops guarantee deg >= 1).
// ---------------------------------------------------------------------------
__global__ void degree_kernel(const uint32_t* __restrict__ adj,
                              float* __restrict__ dinv) {
    int row = blockIdx.x * blockDim.x + threadIdx.x;
    if (row >= N_NODES) return;
    const uint32_t* r = adj + (size_t)row * WPR;
    int deg = 0;
    for (int w = 0; w < WPR; ++w) deg += __popc(r[w]);
    dinv[row] = rsqrtf((float)deg);
}

// ---------------------------------------------------------------------------
// Pre-scale the iterate: xsg = dinv * xin (TDM source for the spmv kernel).
// ---------------------------------------------------------------------------
__global__ void scale_kernel(const float* __restrict__ dinv,
                             const float* __restrict__ xin,
                             float* __restrict__ xsg) {
    int i = blockIdx.x * blockDim.x + threadIdx.x;
    if (i < N_NODES) xsg[i] = dinv[i] * xin[i];
}

// ---------------------------------------------------------------------------
// One diffusion iteration:
//   xs (LDS)   <- xsg via Tensor Data Mover (64 KB contiguous 1-row tile)
//   y_i        = sum_j A[i][j] * xs[j]     (V_WMMA_F32_16X16X4_F32)
//   xout_i     = ALPHA * dinv_i * y_i + (1-ALPHA) * xin_i
//
// 256 blocks x 512 threads. Each block owns 4 tiles (64 rows); each tile is
// computed by KSPLIT=4 waves, each covering a 4096-column K-slice (128 words,
// 1024 WMMAs), partials combined through LDS.
//
// A-fragment (32-bit A 16x4, ISA layout): lane l -> row M = l&15;
//   lanes 0-15 hold K=0,1; lanes 16-31 hold K=2,3.
// B-fragment (4x16): B[k][n] = xs[k0+k] for all n.
// C/D layout: lane 0 holds rows 0-7 (col 0) in VGPRs 0-7; lane 16 rows 8-15.
// ---------------------------------------------------------------------------
__global__ __launch_bounds__(512)
void spmv_wmma_kernel(const uint32_t* __restrict__ adj,
                      const float* __restrict__ dinv,
                      const float* __restrict__ xsg,
                      const float* __restrict__ xin,
                      float* __restrict__ xout) {
    __shared__ float xs[N_NODES];     // 64 KB: normalized input vector
    __shared__ float ysh[16][16];     // per-wave partial row sums

    const int tid  = threadIdx.x;
    const int wv   = tid >> 5;                  // wave in block (0..15)
    const int lane = tid & 31;

    // --- Stage xs into LDS with the Tensor Data Mover (wave 0 issues) ------
    if (wv == 0) {
        uint64_t ga = (uint64_t)(uintptr_t)xsg;     // global source
        uint32_t la = (uint32_t)(uintptr_t)xs;      // LDS destination offset
        u32x4 g0;
        g0.x = 1u;                                  // count=1 (valid user D#)
        g0.y = la;                                  // lds_addr [63:32]
        g0.z = (uint32_t)ga;                        // global_addr[31:0]
        g0.w = (uint32_t)(ga >> 32) | (2u << 30);   // global_addr[56:32]|type=2
        i32x8 g1;
        g1[0] = 0x20000;                            // data_size=2 (4 bytes)
        g1[1] = 0x40000000;                         // tensor_dim0=16384 (lo16)
        g1[2] = 0x00010000;                         // tensor_dim1=1
        g1[3] = 0x40000000;                         // tile_dim0=16384
        g1[4] = 1;                                  // tile_dim1=1
        g1[5] = 16384;                              // tensor_dim0_stride
        g1[6] = 0x40000000;                         // tensor_dim1_stride lo16
        g1[7] = 0;
        i32x4 z4 = {0, 0, 0, 0};                    // groups 2/3: <=2D tensor
        i32x8 z8 = {0, 0, 0, 0, 0, 0, 0, 0};        // trailing group (unused)
        __builtin_amdgcn_tensor_load_to_lds(g0, g1, z4, z4, z8, 0);
        __builtin_amdgcn_s_wait_tensorcnt(0);       // TDM tile landed in LDS
    }
    __syncthreads();

    // --- WMMA accumulation over this wave's K-slice ------------------------
    const int tib     = wv >> 2;                    // tile in block (0..3)
    const int kslice  = wv & 3;                     // K-slice (0..3)
    const int rowbase = (blockIdx.x * 4 + tib) * 16;
    const int m       = lane & 15;                  // my A-matrix row
    const int koff    = (lane >> 4) * 2;            // my K pair offset (0 or 2)
    const uint32_t* rowbits =
        adj + (size_t)(rowbase + m) * WPR + kslice * KWORDS;

    v8f c = {};
    for (int w = 0; w < KWORDS; ++w) {
        uint32_t bits = rowbits[w];
        int kw = (kslice * KWORDS + w) << 5;
#pragma unroll
        for (int s = 0; s < 8; ++s) {
            int b0 = (s << 2) + koff;
            v2f a, b;
            a[0] = (bits >> b0) & 1u ? 1.0f : 0.0f;
            a[1] = (bits >> (b0 + 1)) & 1u ? 1.0f : 0.0f;
            b[0] = xs[kw + b0];
            b[1] = xs[kw + b0 + 1];
            // D = A(16x4) * B(4x16) + C, f32 accumulate
            c = __builtin_amdgcn_wmma_f32_16x16x4_f32(
                    /*neg_a=*/false, a, /*neg_b=*/false, b,
                    /*c_mod=*/(short)0, c, /*reuse_a=*/false, /*reuse_b=*/false);
        }
    }

    // --- Extract partial row sums (column 0 = lane 0 / lane 16) ------------
    if (lane == 0) {
#pragma unroll
        for (int r = 0; r < 8; ++r) ysh[wv][r] = c[r];
    }
    if (lane == 16) {
#pragma unroll
        for (int r = 0; r < 8; ++r) ysh[wv][8 + r] = c[r];
    }
    __syncthreads();

    // --- Combine the 4 K-slices and apply the update -----------------------
    if (tid < 64) {
        int t = tid >> 4;                           // tile in block
        int r = tid & 15;                           // row in tile
        float y = ysh[t * 4 + 0][r] + ysh[t * 4 + 1][r] +
                  ysh[t * 4 + 2][r] + ysh[t * 4 + 3][r];
        int i = blockIdx.x * 64 + t * 16 + r;
        xout[i] = ALPHA * dinv[i] * y + (1.0f - ALPHA) * xin[i];
    }
}

// ---------------------------------------------------------------------------
// Mean BCE-with-logits against thresholded labels; block reduce + atomicAdd.
// ---------------------------------------------------------------------------
__global__ void loss_kernel(const float* __restrict__ emb,
                            const float* __restrict__ xfinal,
                            float* __restrict__ out, int n) {
    __shared__ float red[256];
    int tid = threadIdx.x;
    float acc = 0.0f;
    for (int i = blockIdx.x * blockDim.x + tid; i < n;
         i += gridDim.x * blockDim.x) {
        float e = emb[i];
        float p = (xfinal[i] >= 0.5f) ? 1.0f : 0.0f;
        acc += fmaxf(e, 0.0f) - e * p + log1pf(expf(-fabsf(e)));
    }
    red[tid] = acc;
    __syncthreads();
    for (int s = 128; s > 0; s >>= 1) {
        if (tid < s) red[tid] += red[tid + s];
        __syncthreads();
    }
    if (tid == 0) atomicAdd(out, red[0] * (1.0f / (float)n));
}

// ---------------------------------------------------------------------------
extern "C" void kernel_launch(void* const* d_in, const int* in_sizes, int n_in,
                              void* d_out, int out_size, void* d_ws, size_t ws_size,
                              hipStream_t stream) {
    const float* emb    = (const float*)d_in[0];
    const int*   ei     = (const int*)d_in[1];
    const int*   labels = (const int*)d_in[2];
    float* out = (float*)d_out;

    const int e_total = in_sizes[1] / 2;
    const int n_pos   = in_sizes[2];

    // Workspace layout: [adj bitmask 32 MiB][dinv][x0][x1][xsg]
    uint32_t* adj = (uint32_t*)d_ws;
    float* dinv = (float*)((char*)d_ws + ADJ_WORDS * sizeof(uint32_t));
    float* x0  = dinv + N_NODES;
    float* x1  = x0 + N_NODES;
    float* xsg = x1 + N_NODES;

    init_kernel<<<1024, 256, 0, stream>>>(adj, x0, out);
    scatter_edges_kernel<<<(e_total + 255) / 256, 256, 0, stream>>>(ei, adj, e_total);
    scatter_labels_kernel<<<(n_pos + 255) / 256, 256, 0, stream>>>(labels, x0, n_pos);
    degree_kernel<<<(N_NODES + 255) / 256, 256, 0, stream>>>(adj, dinv);

    float* xin = x0;
    float* xout = x1;
    for (int it = 0; it < N_ITER; ++it) {
        scale_kernel<<<(N_NODES + 255) / 256, 256, 0, stream>>>(dinv, xin, xsg);
        spmv_wmma_kernel<<<256, 512, 0, stream>>>(adj, dinv, xsg, xin, xout);
        float* t = xin; xin = xout; xout = t;
    }
    // after an even number of iterations the result is back in x0 (== xin)
    loss_kernel<<<64, 256, 0, stream>>>(emb, xin, out, N_NODES);
}